// CustomSAGEConv_32598801777055
// MI455X (gfx1250) — compile-verified
//
#include <hip/hip_runtime.h>
#include <hip/hip_bf16.h>

// ---------------------------------------------------------------------------
// CDNA5 (gfx1250) SAGEConv: gather -> segment-mean -> dual GEMM (WMMA fp32)
// -> row L2-normalize.
//
// MI455X reasoning: scatter phase is the bound (~0.82 GB random 512B gathers
// + ~205M fp32 atomics that stay resident in the 192MB L2).  The GEMM phase
// is ~154MB of traffic / 6.6 GFLOP -> memory-bound, so fp32 WMMA
// (V_WMMA_F32_16X16X4_F32) is the right precision: matches reference dtype
// exactly and costs nothing at this arithmetic intensity.  Tiles are staged
// into LDS with GLOBAL_LOAD_ASYNC_TO_LDS_B128 (ASYNCcnt path, no VGPR
// round-trip), and the two GEMMs run as independent WMMA accumulator chains
// to break the D->C RAW serialization.
// ---------------------------------------------------------------------------

typedef __attribute__((ext_vector_type(2))) float v2f;
typedef __attribute__((ext_vector_type(8))) float v8f;

#define DFEAT 128
#define TILE_M 16
#define LDS_PITCH 132   // 128 + 4 pad: row stride 132 dwords -> bank = (4*r + c) % 64, conflict-free

// -------- gfx1250 async global->LDS copy (16 bytes per lane) ---------------
// VDST = per-lane LDS byte address, VADDR = per-lane 64-bit global address.
__device__ __forceinline__ void async_load_b128(unsigned int lds_addr, const void* gaddr) {
    asm volatile("global_load_async_to_lds_b128 %0, %1, off"
                 :: "v"(lds_addr), "v"(gaddr)
                 : "memory");
}
__device__ __forceinline__ void wait_asynccnt0() {
    asm volatile("s_wait_asynccnt 0" ::: "memory");
}

// ---------------------------------------------------------------------------
// Kernel 1: edge scatter.  One wave32 per edge: lane L loads float4 L of the
// source row and issues 4 hardware fp32 atomics into summed[col].  The 51 MB
// accumulator is resident in the 192 MB L2, so atomics stay on-chip.
// ---------------------------------------------------------------------------
__global__ __launch_bounds__(256) void sage_scatter_kernel(
    const float* __restrict__ x,
    const long long* __restrict__ row_idx,
    const long long* __restrict__ col_idx,
    float* __restrict__ summed,
    float* __restrict__ count,
    int num_edges)
{
    const int edge = (int)((blockIdx.x * blockDim.x + threadIdx.x) >> 5);
    const int lane = threadIdx.x & 31;
    if (edge >= num_edges) return;

    const long r = (long)row_idx[edge];
    const long c = (long)col_idx[edge];

    const float4 v = ((const float4*)(x + r * DFEAT))[lane];
    float* dst = summed + c * DFEAT + lane * 4;
    unsafeAtomicAdd(dst + 0, v.x);   // global_atomic_add_f32 (no CAS fallback)
    unsafeAtomicAdd(dst + 1, v.y);
    unsafeAtomicAdd(dst + 2, v.z);
    unsafeAtomicAdd(dst + 3, v.w);
    if (lane == 0) unsafeAtomicAdd(count + c, 1.0f);
}

// ---------------------------------------------------------------------------
// Kernel 2: fused dual-GEMM + L2 normalize.
//   out[m,n] = (summed[m,:]/max(cnt[m],1)) . W_agg[n,:] + x[m,:] . W_lin[n,:]
//   then out[m,:] /= max(||out[m,:]||_2, 1e-12)
//
// Block: 256 threads = 8 wave32.  Block -> 16-row M tile; wave w -> N tile
// [16w, 16w+16).  V_WMMA_F32_16X16X4_F32 operand layouts (ISA 7.12.2):
//   A (16x4):  lane<16: M=lane, {A.x,A.y}=K{0,1};  lane>=16: M=lane-16, K{2,3}
//   B (4x16):  lane<16: N=lane, {B.x,B.y}=K{0,1};  lane>=16: N=lane-16, K{2,3}
//   C/D (16x16): vgpr i: lane<16 -> (M=i, N=lane); lane>=16 -> (M=i+8, N=lane-16)
// The mean-divide is folded into the epilogue: acc row (i + half*8) is scaled
// by inv_cnt[i + half*8] (uniform across lanes of a half), so the raw summed
// tile can be async-copied byte-for-byte.
// ---------------------------------------------------------------------------
__global__ __launch_bounds__(256) void sage_gemm_norm_kernel(
    const float* __restrict__ x,      // [N, 128]
    const float* __restrict__ W_lin,  // [128, 128] row-major
    const float* __restrict__ W_agg,  // [128, 128] row-major
    const float* __restrict__ summed, // [N, 128]
    const float* __restrict__ count,  // [N]
    float* __restrict__ out,          // [N, 128]
    int num_nodes)
{
    __shared__ float xs[TILE_M][LDS_PITCH];   // x tile (async-staged)
    __shared__ float ss[TILE_M][LDS_PITCH];   // raw summed tile (async-staged)
    __shared__ float os[TILE_M][LDS_PITCH];   // output tile (pre-normalize)
    __shared__ float inv_cnt[TILE_M];
    __shared__ float inv_norm[TILE_M];

    const int tid  = threadIdx.x;
    const int wave = tid >> 5;            // 0..7 -> N tile
    const int lane = tid & 31;
    const int half = lane >> 4;           // 0 or 1 (selects K pair)
    const int l15  = lane & 15;
    const long m0  = (long)blockIdx.x * TILE_M;

    // ---- Stage tiles via async global->LDS (ASYNCcnt), 16B per lane ----
    for (int i = tid; i < TILE_M * (DFEAT / 4); i += 256) {
        const int r  = i >> 5;            // i / 32
        const int c4 = i & 31;            // i % 32
        const long row = m0 + r;
        const unsigned int lx = (unsigned int)(size_t)&xs[r][c4 * 4];
        const unsigned int ls = (unsigned int)(size_t)&ss[r][c4 * 4];
        if (row < num_nodes) {
            async_load_b128(lx, x + row * DFEAT + c4 * 4);
            async_load_b128(ls, summed + row * DFEAT + c4 * 4);
        } else {
            *(float4*)&xs[r][c4 * 4] = make_float4(0.f, 0.f, 0.f, 0.f);
            *(float4*)&ss[r][c4 * 4] = make_float4(0.f, 0.f, 0.f, 0.f);
        }
    }
    if (tid < TILE_M) {
        const long row = m0 + tid;
        const float c = (row < num_nodes) ? count[row] : 1.0f;
        inv_cnt[tid] = 1.0f / fmaxf(c, 1.0f);
    }
    wait_asynccnt0();                      // s_wait_asynccnt 0
    __syncthreads();

    // ---- K loop: 32 steps of K=4; two INDEPENDENT WMMA chains ----
    const int n0 = wave * 16;
    v8f accA = {0.f, 0.f, 0.f, 0.f, 0.f, 0.f, 0.f, 0.f};  // summed . W_agg^T
    v8f accX = {0.f, 0.f, 0.f, 0.f, 0.f, 0.f, 0.f, 0.f};  // x . W_lin^T
    for (int k = 0; k < DFEAT; k += 4) {
        const int ka = k + half * 2;
        const v2f a_s = *(const v2f*)&ss[l15][ka];
        const v2f a_x = *(const v2f*)&xs[l15][ka];
        // B[k][n] = W[n][k] (W row-major) -> contiguous float2 along K
        const v2f b_a = *(const v2f*)&W_agg[(long)(n0 + l15) * DFEAT + ka];
        const v2f b_l = *(const v2f*)&W_lin[(long)(n0 + l15) * DFEAT + ka];
        accA = __builtin_amdgcn_wmma_f32_16x16x4_f32(
            false, a_s, false, b_a, (short)0, accA, false, false);
        accX = __builtin_amdgcn_wmma_f32_16x16x4_f32(
            false, a_x, false, b_l, (short)0, accX, false, false);
    }

    // ---- Epilogue: fold mean-divide, spill D tile to LDS ----
#pragma unroll
    for (int i = 0; i < 8; ++i) {
        const int rt = i + half * 8;                // output row within tile
        os[rt][n0 + l15] = accA[i] * inv_cnt[rt] + accX[i];
    }
    __syncthreads();

    // ---- Row L2 norms: 16 threads per row, shfl_xor reduce within wave32 ----
    {
        const int r = tid >> 4;           // 0..15
        const int t = tid & 15;
        float s = 0.f;
#pragma unroll
        for (int j = 0; j < 8; ++j) {
            const float v = os[r][t + 16 * j];
            s += v * v;
        }
        // groups of 16 lanes are contiguous inside a wave32; xor masks < 16
        for (int off = 8; off > 0; off >>= 1) s += __shfl_xor(s, off, 32);
        if (t == 0) inv_norm[r] = 1.0f / fmaxf(sqrtf(s), 1e-12f);
    }
    __syncthreads();

    // ---- Normalize + store ----
    for (int i = tid; i < TILE_M * (DFEAT / 4); i += 256) {
        const int r  = i >> 5;
        const int c4 = i & 31;
        const long row = m0 + r;
        if (row < num_nodes) {
            const float inv = inv_norm[r];
            float4 v = *(float4*)&os[r][c4 * 4];
            v.x *= inv; v.y *= inv; v.z *= inv; v.w *= inv;
            ((float4*)(out + row * DFEAT))[c4] = v;
        }
    }
}

// ---------------------------------------------------------------------------
// Launcher.  Inputs (reference order): x[N,128] f32, W_lin[128,128] f32,
// W_agg[128,128] f32, edge_index[2,E] int64.  Output: [N,128] f32.
// Workspace layout: summed[N*128] f32 | count[N] f32.
// ---------------------------------------------------------------------------
extern "C" void kernel_launch(void* const* d_in, const int* in_sizes, int n_in,
                              void* d_out, int out_size, void* d_ws, size_t ws_size,
                              hipStream_t stream) {
    const float*     x     = (const float*)d_in[0];
    const float*     W_lin = (const float*)d_in[1];
    const float*     W_agg = (const float*)d_in[2];
    const long long* eidx  = (const long long*)d_in[3];   // int64 in reference
    float*           out   = (float*)d_out;

    const int N = in_sizes[0] / DFEAT;       // 100000
    const int E = in_sizes[3] / 2;           // 1600000

    float* summed = (float*)d_ws;
    float* count  = summed + (size_t)N * DFEAT;

    // zero accumulators (memset node is graph-capture safe)
    hipMemsetAsync(d_ws, 0, ((size_t)N * DFEAT + (size_t)N) * sizeof(float), stream);

    // one wave32 per edge -> E/8 blocks of 256 threads
    const int scatter_blocks = (E + 7) / 8;
    sage_scatter_kernel<<<scatter_blocks, 256, 0, stream>>>(
        x, eidx, eidx + E, summed, count, E);

    const int gemm_blocks = (N + TILE_M - 1) / TILE_M;   // 6250
    sage_gemm_norm_kernel<<<gemm_blocks, 256, 0, stream>>>(
        x, W_lin, W_agg, summed, count, out, N);
}